// CognitionNetwork_37151467110481
// MI455X (gfx1250) — compile-verified
//
#include <hip/hip_runtime.h>
#include <hip/hip_bf16.h>
#include <math.h>

// Problem constants (from reference)
#define NN      131072
#define BSEG    2048
#define FF      200      // F
#define F2      400      // 2F
#define F3      600      // 3F  (GEMM K = [q_star | h])
#define F4      800      // 4F  (gates)
#define STEPS   3

typedef float v2f __attribute__((ext_vector_type(2)));
typedef float v8f __attribute__((ext_vector_type(8)));

// ---------------------------------------------------------------------------
// K0: segment ranges via binary search (batch is sorted)
// ---------------------------------------------------------------------------
__global__ void seg_bounds_kernel(const int* __restrict__ batch,
                                  int* __restrict__ seg_s, int* __restrict__ seg_e) {
  int b = blockIdx.x * blockDim.x + threadIdx.x;
  if (b >= BSEG) return;
  auto lb = [&](int key) {
    int lo = 0, hi = NN;
    while (lo < hi) { int mid = (lo + hi) >> 1; if (batch[mid] < key) lo = mid + 1; else hi = mid; }
    return lo;
  };
  seg_s[b] = lb(b);
  seg_e[b] = lb(b + 1);
}

// ---------------------------------------------------------------------------
// K1: A[:,0:400] = q_star ; A[:,400:600] = h0 = seg_sum(cos*x) ; c = 0
// one block per segment
// ---------------------------------------------------------------------------
__global__ void init_kernel(const float* __restrict__ x, const float* __restrict__ cosc,
                            const float* __restrict__ qstar,
                            const int* __restrict__ seg_s, const int* __restrict__ seg_e,
                            float* __restrict__ A, float* __restrict__ c) {
  int b = blockIdx.x;
  int t = threadIdx.x;
  int s = seg_s[b], e = seg_e[b];
  float* Ar = A + (size_t)b * F3;
  const float* q0 = qstar + (size_t)b * F2;
  for (int j = t; j < F2; j += 256) Ar[j] = q0[j];
  if (t < FF) {
    float acc = 0.f;
    for (int i = s; i < e; ++i) acc += cosc[i] * x[(size_t)i * FF + t];
    Ar[F2 + t] = acc;
    c[(size_t)b * FF + t] = 0.f;
  }
}

// ---------------------------------------------------------------------------
// K2: Wcat[g][k] = k<400 ? W_ih[g][k] : W_hh[g][k-400]    (800 x 600)
// ---------------------------------------------------------------------------
__global__ void wcat_kernel(const float* __restrict__ Wih, const float* __restrict__ Whh,
                            float* __restrict__ Wcat) {
  int idx = blockIdx.x * blockDim.x + threadIdx.x;
  if (idx >= F4 * F3) return;
  int g = idx / F3, k = idx - g * F3;
  Wcat[idx] = (k < F2) ? Wih[(size_t)g * F2 + k] : Whh[(size_t)g * FF + (k - F2)];
}

// ---------------------------------------------------------------------------
// K3: gates[2048 x 800] = A[2048 x 600] @ Wcat^T   via V_WMMA_F32_16X16X4_F32
// one wave (32 lanes) per 16x16 output tile; two accumulator chains over K
// ---------------------------------------------------------------------------
__global__ void gemm_gates_wmma(const float* __restrict__ A, const float* __restrict__ W,
                                float* __restrict__ gates) {
  const int lane = threadIdx.x & 31;
  const int wave = (blockIdx.x * blockDim.x + threadIdx.x) >> 5;
  const int tm = wave & 127;         // 128 M-tiles (2048/16)
  const int tn = wave >> 7;          // 50  N-tiles (800/16)
  if (tn >= F4 / 16) return;         // wave-uniform, EXEC stays all-ones

  const int hi   = lane >> 4;        // K half-select
  const int row  = tm * 16 + (lane & 15);     // A row (M)
  const int gcol = tn * 16 + (lane & 15);     // W row (output column N)
  const float* Ap = A + (size_t)row * F3 + 2 * hi;
  const float* Wp = W + (size_t)gcol * F3 + 2 * hi;

  v8f acc0 = {};
  v8f acc1 = {};
  // K = 600 = 75 * 8; each iteration does two independent 16x16x4 WMMAs
  for (int k = 0; k < F3; k += 8) {
    v2f a0 = *(const v2f*)(Ap + k);
    v2f b0 = *(const v2f*)(Wp + k);
    v2f a1 = *(const v2f*)(Ap + k + 4);
    v2f b1 = *(const v2f*)(Wp + k + 4);
    acc0 = __builtin_amdgcn_wmma_f32_16x16x4_f32(false, a0, false, b0,
                                                 (short)0, acc0, false, false);
    acc1 = __builtin_amdgcn_wmma_f32_16x16x4_f32(false, a1, false, b1,
                                                 (short)0, acc1, false, false);
  }

  // D layout: VGPR r -> M = r + 8*hi, N = lane&15
  float* gout = gates + (size_t)(tm * 16 + 8 * hi) * F4 + gcol;
#pragma unroll
  for (int r = 0; r < 8; ++r)
    gout[(size_t)r * F4] = acc0[r] + acc1[r];
}

// ---------------------------------------------------------------------------
// K4: LSTM activations; writes h into A[:,0:200] (=q) and A[:,400:600] (=h)
// ---------------------------------------------------------------------------
__device__ __forceinline__ float sigm(float v) { return 1.f / (1.f + expf(-v)); }

__global__ void lstm_act_kernel(const float* __restrict__ gates,
                                const float* __restrict__ bih, const float* __restrict__ bhh,
                                float* __restrict__ c, float* __restrict__ A) {
  int idx = blockIdx.x * blockDim.x + threadIdx.x;
  if (idx >= BSEG * FF) return;
  int b = idx / FF, f = idx - b * FF;
  const float* gr = gates + (size_t)b * F4;
  float gi = gr[f]          + bih[f]          + bhh[f];
  float gf = gr[FF + f]     + bih[FF + f]     + bhh[FF + f];
  float gg = gr[2 * FF + f] + bih[2 * FF + f] + bhh[2 * FF + f];
  float go = gr[3 * FF + f] + bih[3 * FF + f] + bhh[3 * FF + f];
  float cn = sigm(gf) * c[idx] + sigm(gi) * tanhf(gg);
  float hn = sigm(go) * tanhf(cn);
  c[idx] = cn;
  float* Ar = A + (size_t)b * F3;
  Ar[f]      = hn;   // q (first half of next q_star)
  Ar[F2 + f] = hn;   // h fed to next LSTM cell
}

// ---------------------------------------------------------------------------
// K5: per-segment attention: e = x.q ; softmax over segment ; r = sum(a*x)
// one block (8 wave32) per segment; writes r into A[:,200:400]
// ---------------------------------------------------------------------------
__global__ void attn_kernel(const float* __restrict__ x,
                            const int* __restrict__ seg_s, const int* __restrict__ seg_e,
                            float* __restrict__ A, float* __restrict__ e_ws) {
  const int b = blockIdx.x;
  const int t = threadIdx.x;
  const int s = seg_s[b], e = seg_e[b];
  const int wave = t >> 5, lane = t & 31;

  __shared__ float q_s[FF];
  __shared__ float red[8];
  __shared__ float m_s, inv_s;

  float* Ar = A + (size_t)b * F3;
  if (t < FF) q_s[t] = Ar[t];          // q = h_new
  __syncthreads();

  // ---- phase A: dots + running max ----
  float lmax = -3.4e38f;
  for (int i = s + wave; i < e; i += 8) {
    const float* xr = x + (size_t)i * FF;
    float d = 0.f;
#pragma unroll
    for (int j = 0; j < 6; ++j) d += xr[lane + 32 * j] * q_s[lane + 32 * j];
    if (lane < 8) d += xr[192 + lane] * q_s[192 + lane];
#pragma unroll
    for (int off = 16; off >= 1; off >>= 1) d += __shfl_xor(d, off, 32);
    if (lane == 0) e_ws[i] = d;
    lmax = fmaxf(lmax, d);
  }
  if (lane == 0) red[wave] = lmax;
  __threadfence_block();
  __syncthreads();
  if (t == 0) {
    float m = -3.4e38f;
    for (int w = 0; w < 8; ++w) m = fmaxf(m, red[w]);
    m_s = m;
  }
  __syncthreads();
  const float m = m_s;

  // ---- phase B: denom, then normalize in place ----
  float ls = 0.f;
  for (int i = s + t; i < e; i += 256) ls += expf(e_ws[i] - m);
#pragma unroll
  for (int off = 16; off >= 1; off >>= 1) ls += __shfl_xor(ls, off, 32);
  if (lane == 0) red[wave] = ls;
  __syncthreads();
  if (t == 0) {
    float den = 0.f;
    for (int w = 0; w < 8; ++w) den += red[w];
    inv_s = (den > 0.f) ? 1.f / den : 0.f;
  }
  __syncthreads();
  const float inv = inv_s;
  for (int i = s + t; i < e; i += 256) e_ws[i] = expf(e_ws[i] - m) * inv;
  __threadfence_block();
  __syncthreads();

  // ---- phase C: r[b][f] = sum a_i * x[i][f] ----
  if (t < FF) {
    float acc = 0.f;
    for (int i = s; i < e; ++i) acc += e_ws[i] * x[(size_t)i * FF + t];
    Ar[FF + t] = acc;   // r -> second half of q_star
  }
}

// ---------------------------------------------------------------------------
// K6: out[b][0:400] = A[b][0:400]
// ---------------------------------------------------------------------------
__global__ void out_copy_kernel(const float* __restrict__ A, float* __restrict__ out) {
  int idx = blockIdx.x * blockDim.x + threadIdx.x;
  if (idx >= BSEG * F2) return;
  int b = idx / F2, j = idx - b * F2;
  out[idx] = A[(size_t)b * F3 + j];
}

// ---------------------------------------------------------------------------
extern "C" void kernel_launch(void* const* d_in, const int* in_sizes, int n_in,
                              void* d_out, int out_size, void* d_ws, size_t ws_size,
                              hipStream_t stream) {
  const float* x     = (const float*)d_in[0];
  const int*   batch = (const int*)  d_in[1];
  const float* cosc  = (const float*)d_in[2];
  const float* qstar = (const float*)d_in[3];
  const float* Wih   = (const float*)d_in[4];
  const float* Whh   = (const float*)d_in[5];
  const float* bih   = (const float*)d_in[6];
  const float* bhh   = (const float*)d_in[7];
  float* out = (float*)d_out;

  // workspace carve-up (~15.6 MB)
  char* p = (char*)d_ws;
  int*   seg_s = (int*)p;   p += (size_t)BSEG * 4;
  int*   seg_e = (int*)p;   p += (size_t)BSEG * 4;
  float* A     = (float*)p; p += (size_t)BSEG * F3 * 4;
  float* gates = (float*)p; p += (size_t)BSEG * F4 * 4;
  float* c     = (float*)p; p += (size_t)BSEG * FF * 4;
  float* Wcat  = (float*)p; p += (size_t)F4 * F3 * 4;
  float* e_ws  = (float*)p; p += (size_t)NN * 4;
  (void)in_sizes; (void)n_in; (void)out_size; (void)ws_size;

  seg_bounds_kernel<<<(BSEG + 255) / 256, 256, 0, stream>>>(batch, seg_s, seg_e);
  wcat_kernel<<<(F4 * F3 + 255) / 256, 256, 0, stream>>>(Wih, Whh, Wcat);
  init_kernel<<<BSEG, 256, 0, stream>>>(x, cosc, qstar, seg_s, seg_e, A, c);

  // GEMM grid: 128*50 = 6400 waves -> 800 blocks of 256 threads (8 wave32)
  for (int s = 0; s < STEPS; ++s) {
    gemm_gates_wmma<<<800, 256, 0, stream>>>(A, Wcat, gates);
    lstm_act_kernel<<<(BSEG * FF + 255) / 256, 256, 0, stream>>>(gates, bih, bhh, c, A);
    attn_kernel<<<BSEG, 256, 0, stream>>>(x, seg_s, seg_e, A, e_ws);
  }
  out_copy_kernel<<<(BSEG * F2 + 255) / 256, 256, 0, stream>>>(A, out);
}